// PointerGenerator_90340342104741
// MI455X (gfx1250) — compile-verified
//
#include <hip/hip_runtime.h>
#include <math.h>

typedef __attribute__((ext_vector_type(2))) float v2f;
typedef __attribute__((ext_vector_type(8))) float v8f;
typedef __attribute__((ext_vector_type(4))) float f4;

#define B_   512
#define T_   400
#define V_   32000
#define ENC_ 1024
#define HID_ 1024
#define EMB_ 512

// ---------------------------------------------------------------------------
// Kernel 1: p_gen = sigmoid(context@w_c + state@w_s + emb@w_y + bias)
// One wave32 per 16 batch rows. Uses V_WMMA_F32_16X16X4_F32 (fp32 path, no
// precision loss). A-fragment layout (32-bit A, 16x4): lanes 0-15 hold
// (M=lane, K=0,1) in v0,v1; lanes 16-31 hold (M=lane-16, K=2,3).
// B-fragment: row K broadcast across all 16 N-lanes -> every column of D is
// the same dot product, so lane 0 (cols of rows 0..7) / lane 16 (rows 8..15)
// can write the result directly.
// ---------------------------------------------------------------------------
__global__ __launch_bounds__(32) void pgen_wmma_kernel(
    const float* __restrict__ context, const float* __restrict__ state,
    const float* __restrict__ emb,
    const float* __restrict__ w_c, const float* __restrict__ w_s,
    const float* __restrict__ w_y, const float* __restrict__ bias,
    float* __restrict__ p_gen)
{
    const int lane      = threadIdx.x;        // 0..31, full wave, EXEC all-1s
    const int tile      = blockIdx.x;         // 0..31  (B/16 tiles)
    const int rowInTile = lane & 15;
    const int hiHalf    = lane >> 4;          // 0 -> K{0,1}, 1 -> K{2,3}
    const int row       = tile * 16 + rowInTile;

    v8f acc = {};

    auto mac_segment = [&](const float* __restrict__ X,
                           const float* __restrict__ w, int D) {
        const float* xrow = X + (size_t)row * (size_t)D;
        for (int k = 0; k < D; k += 4) {
            const int k0 = k + hiHalf * 2;
            v2f a;
            a.x = xrow[k0];
            a.y = xrow[k0 + 1];
            v2f bf;
            bf.x = w[k0];          // broadcast across N lanes of this K-half
            bf.y = w[k0 + 1];
            acc = __builtin_amdgcn_wmma_f32_16x16x4_f32(
                /*neg_a=*/false, a, /*neg_b=*/false, bf,
                /*c_mod=*/(short)0, acc, /*reuse_a=*/false, /*reuse_b=*/false);
        }
    };

    mac_segment(context, w_c, ENC_);
    mac_segment(state,   w_s, HID_);
    mac_segment(emb,     w_y, EMB_);

    const float bv = bias[0];
    // C/D layout: lane 0, VGPR j -> (M=j, N=0); lane 16, VGPR j -> (M=8+j, N=0)
    if (rowInTile == 0) {
        const int mBase = tile * 16 + hiHalf * 8;
#pragma unroll
        for (int j = 0; j < 8; ++j) {
            const float logit = acc[j] + bv;
            p_gen[mBase + j] = 1.0f / (1.0f + __expf(-logit));
        }
    }
}

// ---------------------------------------------------------------------------
// Kernel 2: out[b,v] = p_gen[b] * vocab_dist[b,v]   (bandwidth-critical pass)
// float4-vectorized; V=32000 divisible by 4 so vectors never cross rows.
// ---------------------------------------------------------------------------
__global__ __launch_bounds__(256) void scale_vocab_kernel(
    const float* __restrict__ vocab, const float* __restrict__ p_gen,
    float* __restrict__ out)
{
    const size_t n4 = (size_t)B_ * (size_t)V_ / 4;
    size_t i = (size_t)blockIdx.x * blockDim.x + threadIdx.x;
    if (i >= n4) return;

    const int b = (int)((i * 4) / V_);
    const float pg = p_gen[b];

    f4 vd = ((const f4*)vocab)[i];
    f4 o;
    o.x = pg * vd.x;
    o.y = pg * vd.y;
    o.z = pg * vd.z;
    o.w = pg * vd.w;
    ((f4*)out)[i] = o;
}

// ---------------------------------------------------------------------------
// Kernel 3: masked scatter-add of the copy distribution.
// out[b, src_ids[b,t]] += (1 - p_gen[b]) * attn[b,t]   if src_ids < V
// (reference adds 0 for ids >= V, so skipping is exact.)
// ---------------------------------------------------------------------------
__global__ __launch_bounds__(256) void scatter_copy_kernel(
    const float* __restrict__ attn, const int* __restrict__ src_ids,
    const float* __restrict__ p_gen, float* __restrict__ out)
{
    const int n = B_ * T_;
    int i = blockIdx.x * blockDim.x + threadIdx.x;
    if (i >= n) return;

    const int b  = i / T_;
    const int id = src_ids[i];
    if (id < V_) {
        const float pc = (1.0f - p_gen[b]) * attn[i];
        atomicAdd(out + (size_t)b * (size_t)V_ + (size_t)id, pc);
    }
}

// ---------------------------------------------------------------------------
// Launch. Input order (setup_inputs): 0 vocab_dist, 1 attn_dist, 2 context,
// 3 state, 4 emb, 5 src_ids, 6 vocab_size(scalar), 7 w_c, 8 w_s, 9 w_y, 10 b
// ---------------------------------------------------------------------------
extern "C" void kernel_launch(void* const* d_in, const int* in_sizes, int n_in,
                              void* d_out, int out_size, void* d_ws, size_t ws_size,
                              hipStream_t stream)
{
    const float* vocab_dist = (const float*)d_in[0];
    const float* attn_dist  = (const float*)d_in[1];
    const float* context    = (const float*)d_in[2];
    const float* state      = (const float*)d_in[3];
    const float* emb        = (const float*)d_in[4];
    const int*   src_ids    = (const int*)  d_in[5];
    const float* w_c        = (const float*)d_in[7];
    const float* w_s        = (const float*)d_in[8];
    const float* w_y        = (const float*)d_in[9];
    const float* bias       = (const float*)d_in[10];

    float* out   = (float*)d_out;
    float* p_gen = (float*)d_ws;           // B_ floats of scratch

    // 1) p_gen via fp32 WMMA GEMV: 32 tiles of 16 rows, one wave32 each.
    pgen_wmma_kernel<<<B_ / 16, 32, 0, stream>>>(
        context, state, emb, w_c, w_s, w_y, bias, p_gen);

    // 2) scale the vocab distribution (dominant 131 MB stream).
    const size_t n4 = (size_t)B_ * (size_t)V_ / 4;
    const int blocks2 = (int)((n4 + 255) / 256);
    scale_vocab_kernel<<<blocks2, 256, 0, stream>>>(vocab_dist, p_gen, out);

    // 3) scatter-add the copy distribution.
    const int n3 = B_ * T_;
    const int blocks3 = (n3 + 255) / 256;
    scatter_copy_kernel<<<blocks3, 256, 0, stream>>>(attn_dist, src_ids, p_gen, out);
}